// GCN_special_37194416783907
// MI455X (gfx1250) — compile-verified
//
#include <hip/hip_runtime.h>
#include <math.h>

typedef float v2f __attribute__((ext_vector_type(2)));
typedef float v8f __attribute__((ext_vector_type(8)));

// ---------------------------------------------------------------- utilities
__global__ void k_fill(float* __restrict__ p, float v, int n) {
    int i = blockIdx.x * blockDim.x + threadIdx.x;
    if (i < n) p[i] = v;
}

__global__ void k_deg_accum(const int* __restrict__ dst, float* __restrict__ deg, int E) {
    int i = blockIdx.x * blockDim.x + threadIdx.x;
    if (i < E) unsafeAtomicAdd(&deg[dst[i]], 1.0f);   // global_atomic_add_f32, no return
}

__global__ void k_rsqrt_inplace(float* __restrict__ d, int n) {
    int i = blockIdx.x * blockDim.x + threadIdx.x;
    if (i < n) d[i] = rsqrtf(d[i]);                   // deg >= 1 (self-loop), always valid
}

// Zero-pad weight matrix [K x Ncol] -> [K x Np] (Np = Ncol rounded up to 16).
__global__ void k_pad_weight(const float* __restrict__ W, float* __restrict__ Wp,
                             int K, int Ncol, int Np) {
    int idx = blockIdx.x * blockDim.x + threadIdx.x;
    if (idx >= K * Np) return;
    int k = idx / Np, c = idx - k * Np;
    Wp[idx] = (c < Ncol) ? W[(size_t)k * Ncol + c] : 0.0f;
}

// ---------------------------------------------------------------- WMMA GEMM
// C[M x Np] = A[M x K] * B[K x Np], row-major f32.  REQUIRES: M%16==0, Np%16==0.
// One wave computes one 16x16 tile via V_WMMA_F32_16X16X4_F32; no predication
// anywhere in the loop, so EXEC stays all-1s and loads are unconditional.
__global__ __launch_bounds__(128)
void k_wmma_gemm_f32(const float* __restrict__ A, const float* __restrict__ B,
                     float* __restrict__ C, int K, int Np) {
    int mt = blockIdx.x;
    int nt = blockIdx.y * blockDim.y + threadIdx.y;

    int lane = threadIdx.x;                           // 0..31 (wave32)
    int half = lane >> 4;                             // K-pair select
    int l16  = lane & 15;

    const float* __restrict__ arow = A + (size_t)(mt * 16 + l16) * (size_t)K;
    int col = nt * 16 + l16;

    v8f acc = {0.f, 0.f, 0.f, 0.f, 0.f, 0.f, 0.f, 0.f};

    #pragma unroll 4
    for (int k = 0; k < K; k += 4) {
        int ka = k + 2 * half;                        // lanes 0-15: K=k,k+1; 16-31: K=k+2,k+3
        v2f a, b;
        a.x = arow[ka];
        a.y = arow[ka + 1];
        b.x = B[(size_t)ka * Np + col];
        b.y = B[(size_t)(ka + 1) * Np + col];
        acc = __builtin_amdgcn_wmma_f32_16x16x4_f32(
            /*neg_a=*/false, a, /*neg_b=*/false, b,
            /*c_mod=*/(short)0, acc, /*reuse_a=*/false, /*reuse_b=*/false);
    }

    #pragma unroll
    for (int r = 0; r < 8; ++r)                       // VGPR r -> rows r (lanes<16) / r+8
        C[(size_t)(mt * 16 + 8 * half + r) * Np + col] = acc[r];
}

// ------------------------------------------------------- aggregation kernels
// Seed accumulator with self-loop term: agg[i, 0..F) = dinv[i]^2 * H[i*Hstride + f]
__global__ void k_init_agg(const float* __restrict__ H, const float* __restrict__ dinv,
                           float* __restrict__ agg, int N, int F, int Hstride) {
    int idx = blockIdx.x * blockDim.x + threadIdx.x;
    if (idx >= N * F) return;
    int i = idx / F, f = idx - i * F;
    float di = dinv[i];
    agg[idx] = di * di * H[(size_t)i * Hstride + f];
}

// 128-feature scatter: one wave per edge, 4 features per lane.
__global__ __launch_bounds__(256)
void k_scatter128(const int* __restrict__ src, const int* __restrict__ dst,
                  const float* __restrict__ dinv, const float* __restrict__ H,
                  float* __restrict__ agg, int E) {
    int wid  = (blockIdx.x * blockDim.x + threadIdx.x) >> 5;
    int lane = threadIdx.x & 31;
    if (wid >= E) return;
    int s = src[wid], d = dst[wid];
    float nrm = dinv[s] * dinv[d];
    const float* hs = H   + (size_t)s * 128;
    float*       ad = agg + (size_t)d * 128;
    #pragma unroll
    for (int j = 0; j < 4; ++j) {
        int f = lane + 32 * j;
        unsafeAtomicAdd(&ad[f], nrm * hs[f]);
    }
}

// (edge, feature)-flattened scatter; H has row stride Hstride, agg compact F.
__global__ void k_scatter_feat(const int* __restrict__ src, const int* __restrict__ dst,
                               const float* __restrict__ dinv, const float* __restrict__ H,
                               float* __restrict__ agg, int E, int F, int Hstride) {
    int idx = blockIdx.x * blockDim.x + threadIdx.x;
    if (idx >= E * F) return;
    int e = idx / F;
    int f = idx - e * F;
    int s = src[e], d = dst[e];
    float nrm = dinv[s] * dinv[d];
    unsafeAtomicAdd(&agg[(size_t)d * F + f], nrm * H[(size_t)s * Hstride + f]);
}

__global__ void k_bias_relu(float* __restrict__ h, const float* __restrict__ b,
                            int total, int F) {
    int idx = blockIdx.x * blockDim.x + threadIdx.x;
    if (idx < total) h[idx] = fmaxf(h[idx] + b[idx % F], 0.0f);
}

// Fused bias + ReLU + log_softmax over F<=64 features: one wave per row.
__global__ __launch_bounds__(256)
void k_bias_relu_logsoftmax(const float* __restrict__ agg, const float* __restrict__ b,
                            float* __restrict__ out, int N, int F) {
    int row  = blockIdx.x * (blockDim.x >> 5) + (threadIdx.x >> 5);
    int lane = threadIdx.x & 31;
    if (row >= N) return;
    const float* a = agg + (size_t)row * F;
    int f0 = lane, f1 = lane + 32;
    float z0 = (f0 < F) ? fmaxf(a[f0] + b[f0], 0.0f) : -INFINITY;
    float z1 = (f1 < F) ? fmaxf(a[f1] + b[f1], 0.0f) : -INFINITY;
    float m = fmaxf(z0, z1);
    #pragma unroll
    for (int off = 16; off > 0; off >>= 1) m = fmaxf(m, __shfl_xor(m, off, 32));
    float s = ((f0 < F) ? __expf(z0 - m) : 0.0f) + ((f1 < F) ? __expf(z1 - m) : 0.0f);
    #pragma unroll
    for (int off = 16; off > 0; off >>= 1) s += __shfl_xor(s, off, 32);
    float lse = m + __logf(s);
    float* o = out + (size_t)row * F;
    if (f0 < F) o[f0] = z0 - lse;
    if (f1 < F) o[f1] = z1 - lse;
}

// ---------------------------------------------------------------- launcher
extern "C" void kernel_launch(void* const* d_in, const int* in_sizes, int n_in,
                              void* d_out, int out_size, void* d_ws, size_t ws_size,
                              hipStream_t stream) {
    const float* x  = (const float*)d_in[0];
    const int*   ei = (const int*)  d_in[1];
    const float* W1 = (const float*)d_in[2];
    const float* b1 = (const float*)d_in[3];
    const float* W2 = (const float*)d_in[4];
    const float* b2 = (const float*)d_in[5];
    float* out = (float*)d_out;

    const int DH   = in_sizes[3];            // 128
    const int DIN  = in_sizes[2] / DH;       // 128
    const int DOUT = in_sizes[5];            // 40
    const int N    = in_sizes[0] / DIN;      // 50000
    const int E    = in_sizes[1] / 2;        // 800000
    const int DOP  = (DOUT + 15) & ~15;      // 48 (padded layer-2 width)
    const int* src = ei;
    const int* dst = ei + E;

    // workspace carve-out
    char* ws = (char*)d_ws;
    size_t off = 0;
    auto carve = [&](size_t bytes) -> char* {
        char* p = ws + off;
        off += (bytes + 255) & ~(size_t)255;
        return p;
    };
    float* dinv = (float*)carve((size_t)N * 4);              // deg -> dinv in place
    float* H1   = (float*)carve((size_t)N * DH * 4);         // X @ W1
    float* agg1 = (float*)carve((size_t)N * DH * 4);         // layer-1 aggregate / hidden
    float* W2p  = (float*)carve((size_t)DH * DOP * 4);       // zero-padded W2 (128x48)
    float* H2p  = (float*)carve((size_t)N * DOP * 4);        // hidden @ W2p (N x 48)
    float* agg2 = (float*)carve((size_t)N * DOUT * 4);       // layer-2 aggregate (compact)

    const int B = 256;
    auto cdiv = [](int a, int b) { return (a + b - 1) / b; };

    // --- normalization: deg (with self-loop) -> dinv = rsqrt(deg)
    k_fill<<<cdiv(N, B), B, 0, stream>>>(dinv, 1.0f, N);
    k_deg_accum<<<cdiv(E, B), B, 0, stream>>>(dst, dinv, E);
    k_rsqrt_inplace<<<cdiv(N, B), B, 0, stream>>>(dinv, N);

    // --- layer 1: GEMM (f32 WMMA, DH=128 already 16-aligned), seed, scatter, bias+ReLU
    {
        dim3 grid(N / 16, (DH / 16 + 3) / 4), blk(32, 4);
        k_wmma_gemm_f32<<<grid, blk, 0, stream>>>(x, W1, H1, DIN, DH);
    }
    k_init_agg<<<cdiv(N * DH, B), B, 0, stream>>>(H1, dinv, agg1, N, DH, DH);
    k_scatter128<<<cdiv(E * 32, B), B, 0, stream>>>(src, dst, dinv, H1, agg1, E);
    k_bias_relu<<<cdiv(N * DH, B), B, 0, stream>>>(agg1, b1, N * DH, DH);

    // --- layer 2: pad W2 -> 16-aligned, GEMM, seed, scatter, fused epilogue -> d_out
    k_pad_weight<<<cdiv(DH * DOP, B), B, 0, stream>>>(W2, W2p, DH, DOUT, DOP);
    {
        dim3 grid(N / 16, 1), blk(32, DOP / 16);             // 3 waves/block, 48 cols
        k_wmma_gemm_f32<<<grid, blk, 0, stream>>>(agg1, W2p, H2p, DH, DOP);
    }
    k_init_agg<<<cdiv(N * DOUT, B), B, 0, stream>>>(H2p, dinv, agg2, N, DOUT, DOP);
    k_scatter_feat<<<cdiv(E * DOUT, B), B, 0, stream>>>(src, dst, dinv, H2p, agg2, E, DOUT, DOP);
    k_bias_relu_logsoftmax<<<cdiv(N, B / 32), B, 0, stream>>>(agg2, b2, out, N, DOUT);
}